// RPN_23845658428417
// MI455X (gfx1250) — compile-verified
//
#include <hip/hip_runtime.h>
#include <hip/hip_bf16.h>

typedef __attribute__((ext_vector_type(2))) float v2f;
typedef __attribute__((ext_vector_type(8))) float v8f;

#define NB    4
#define CC    256
#define HH    128
#define WW    192
#define HW    (HH*WW)        // 24576
#define NANCH (HW*3)         // 73728
#define NPAD  131072         // 2^17 for bitonic
#define NBOX  6000
#define NWRD  94             // ceil(6000/64)
#define POSTK 1000
#define NEGV  (-1.0e9f)
#define SCALE_CLAMP 4.135166556742356f   // log(1000/16)

// ---------------------------------------------------------------------------
// CDNA5 async memory->LDS copy (GLOBAL_LOAD_ASYNC_TO_LDS_B32, ASYNCcnt).
// Builtin takes int-typed pointees: (as(1) src, as(3) dst, imm offset, imm cpol).
// Guarded with a sync fallback.
// ---------------------------------------------------------------------------
#define AS1 __attribute__((address_space(1)))
#define AS3 __attribute__((address_space(3)))

#if defined(__has_builtin)
#  if __has_builtin(__builtin_amdgcn_global_load_async_to_lds_b32)
#    define HAVE_ASYNC_LDS 1
#  else
#    define HAVE_ASYNC_LDS 0
#  endif
#else
#  define HAVE_ASYNC_LDS 0
#endif

#if HAVE_ASYNC_LDS
__device__ __forceinline__ void async_cp_b32(const void* g, void* l) {
  __builtin_amdgcn_global_load_async_to_lds_b32(
      (AS1 int*)(unsigned long long)g,
      (AS3 int*)(unsigned int)(unsigned long long)l, 0, 0);
}
__device__ __forceinline__ void async_wait0() {
#  if __has_builtin(__builtin_amdgcn_s_wait_asynccnt)
  __builtin_amdgcn_s_wait_asynccnt(0);
#  else
  asm volatile("s_wait_asynccnt 0x0" ::: "memory");
#  endif
}
#endif

// ---------------------------------------------------------------------------
// 3x3 conv + ReLU as implicit GEMM with V_WMMA_F32_16X16X4_F32.
// K ordered (ky,kx,ci). Block = 256 threads (8 waves), tile 64co x 64sp.
// Double-buffered LDS staging: chunk c+1 is copied (async on CDNA5 path)
// while chunk c runs its 16 WMMAs; one barrier per chunk.
// ---------------------------------------------------------------------------
__global__ __launch_bounds__(256) void conv3_wmma(
    const float* __restrict__ feat, const float* __restrict__ wconv,
    const float* __restrict__ bconv, float* __restrict__ t)
{
  __shared__ float As[2][64*34];   // As[buf][co][kk], pad 34 -> 8B-aligned b64 frags
  __shared__ float Bs[2][64*34];   // Bs[buf][sp][kk]
  const int n   = blockIdx.z;
  const int co0 = blockIdx.y * 64;
  const int p0  = blockIdx.x * 64;          // 64 consecutive x in one row (192%64==0)
  const int y0  = p0 / WW, x0 = p0 % WW;
  const int tid = threadIdx.x;
  const int wave = tid >> 5, lane = tid & 31;
  const int m = lane & 15, half = lane >> 4;
  const int cosub = (wave & 3) * 16;        // co sub-tile of this wave
  const int spp   = (wave >> 2) * 32;       // wave owns sp [spp,spp+16) and [spp+16,spp+32)

  v8f acc0 = {}; v8f acc1 = {};

  // stage chunk c (c = kp*8 + ci_chunk) into buffer buf
  auto stage = [&](int c, int buf) {
    const int kp = c >> 3, cb = (c & 7) << 5;
    const int ky = kp / 3, kx = kp % 3;
    const int yy = y0 + ky - 1;
    float* Ab = As[buf];
    float* Bb = Bs[buf];
    // A: 64 co x 32 ci   (w_conv flat: (co*256+ci)*9 + kp)
    #pragma unroll
    for (int i = tid; i < 64*32; i += 256) {
      int kk = i & 31, co = i >> 5;
      const float* src = &wconv[((co0 + co)*CC + (cb + kk))*9 + kp];
#if HAVE_ASYNC_LDS
      async_cp_b32(src, &Ab[co*34 + kk]);
#else
      Ab[co*34 + kk] = *src;
#endif
    }
    // B: 32 ci x 64 sp (transposed into Bs[sp][kk]); zero-pad borders via ds_store
    #pragma unroll
    for (int i = tid; i < 64*32; i += 256) {
      int sp = i & 63, kk = i >> 6;
      int xx = x0 + sp + kx - 1;
      bool in = (xx >= 0) && (xx < WW) && (yy >= 0) && (yy < HH);
#if HAVE_ASYNC_LDS
      if (in) async_cp_b32(&feat[((n*CC + cb + kk)*HH + yy)*WW + xx], &Bb[sp*34 + kk]);
      else    Bb[sp*34 + kk] = 0.f;
#else
      Bb[sp*34 + kk] = in ? feat[((n*CC + cb + kk)*HH + yy)*WW + xx] : 0.f;
#endif
    }
  };

  stage(0, 0);
#if HAVE_ASYNC_LDS
  async_wait0();
#endif
  __syncthreads();

  for (int c = 0; c < 72; ++c) {
    const int buf = c & 1;
    if (c + 1 < 72) stage(c + 1, buf ^ 1);      // overlap next-chunk DMA with WMMA
    const float* Ab = As[buf];
    const float* Bb = Bs[buf];
    #pragma unroll
    for (int kk = 0; kk < 32; kk += 4) {
      int kA = kk + 2*half;                     // ISA 16x4 / 4x16 f32 layout
      v2f a  = *(const v2f*)&Ab[(cosub + m)*34 + kA];
      v2f b0 = *(const v2f*)&Bb[(spp + m)*34 + kA];
      v2f b1 = *(const v2f*)&Bb[(spp + 16 + m)*34 + kA];
      acc0 = __builtin_amdgcn_wmma_f32_16x16x4_f32(false, a, false, b0, (short)0, acc0, false, false);
      acc1 = __builtin_amdgcn_wmma_f32_16x16x4_f32(false, a, false, b1, (short)0, acc1, false, false);
    }
#if HAVE_ASYNC_LDS
    async_wait0();
#endif
    __syncthreads();
  }

  // epilogue: C/D layout — VGPR r holds M = r + 8*half, N = lane%16
  #pragma unroll
  for (int r = 0; r < 8; ++r) {
    int co = co0 + cosub + r + 8*half;
    float bv = bconv[co];
    int p = p0 + spp + m;
    float v0 = acc0[r] + bv; v0 = v0 > 0.f ? v0 : 0.f;
    t[(size_t)(n*CC + co)*HW + p] = v0;
    float v1 = acc1[r] + bv; v1 = v1 > 0.f ? v1 : 0.f;
    t[(size_t)(n*CC + co)*HW + p + 16] = v1;
  }
}

// ---------------------------------------------------------------------------
// 1x1 heads: combined [16 x 256] weight matrix (rows 0-2 obj, 3-14 delta, 15 pad)
// one wave per 16-spatial tile, f32 WMMA over K=256.
// ---------------------------------------------------------------------------
__global__ __launch_bounds__(256) void heads_wmma(
    const float* __restrict__ t, const float* __restrict__ wobj,
    const float* __restrict__ bobj, const float* __restrict__ wdel,
    const float* __restrict__ bdel, float* __restrict__ S, float* __restrict__ D)
{
  __shared__ float Wh[16*258];
  const int n = blockIdx.y;
  const int tid = threadIdx.x;
  for (int i = tid; i < 16*256; i += 256) {
    int co = i >> 8, ci = i & 255;
    const float* src = (co < 3)  ? &wobj[co*256 + ci]
                     : (co < 15) ? &wdel[(co - 3)*256 + ci] : nullptr;
#if HAVE_ASYNC_LDS
    if (src) async_cp_b32(src, &Wh[co*258 + ci]);
    else     Wh[co*258 + ci] = 0.f;
#else
    Wh[co*258 + ci] = src ? *src : 0.f;
#endif
  }
#if HAVE_ASYNC_LDS
  async_wait0();
#endif
  __syncthreads();
  const int wave = tid >> 5, lane = tid & 31, m = lane & 15, half = lane >> 4;
  const int p = blockIdx.x*128 + wave*16 + m;
  const float* tb = t + (size_t)n*CC*HW;
  v8f acc = {};
  #pragma unroll 4
  for (int ci = 0; ci < 256; ci += 4) {
    int kA = ci + 2*half;
    v2f a = *(const v2f*)&Wh[m*258 + kA];
    v2f b;
    b.x = tb[(size_t)kA*HW + p];
    b.y = tb[(size_t)(kA + 1)*HW + p];
    acc = __builtin_amdgcn_wmma_f32_16x16x4_f32(false, a, false, b, (short)0, acc, false, false);
  }
  #pragma unroll
  for (int r = 0; r < 8; ++r) {
    int M = r + 8*half;
    float v = acc[r];
    if (M < 3) {
      S[(size_t)n*NANCH + p*3 + M] = v + bobj[M];           // logits: [n][p*3+a]
    } else if (M < 15) {
      int d = M - 3, a = d >> 2, c = d & 3;                  // deltas: [n][p*3+a][c]
      D[((size_t)n*NANCH + p*3 + a)*4 + c] = v + bdel[d];
    }
  }
}

// ---------------------------------------------------------------------------
// sortable keys: monotone-f32 bits in hi 32, ~index in lo 32 (stable tiebreak)
// ---------------------------------------------------------------------------
__global__ void make_keys(const float* __restrict__ S, unsigned long long* __restrict__ keys)
{
  int gi = blockIdx.x*blockDim.x + threadIdx.x;
  int n = gi >> 17, j = gi & (NPAD - 1);
  if (n >= NB) return;
  unsigned long long k = 0ull;
  if (j < NANCH) {
    unsigned int b = __float_as_uint(S[(size_t)n*NANCH + j]);
    unsigned int s = b ^ ((b >> 31) ? 0xFFFFFFFFu : 0x80000000u);
    k = ((unsigned long long)s << 32) | (unsigned int)(~j);
  }
  keys[((size_t)n << 17) + j] = k;
}

// in-block bitonic sort (descending) of 2^17 keys, one block per batch
__global__ __launch_bounds__(1024) void bitonic_sort(unsigned long long* __restrict__ keys)
{
  unsigned long long* a = keys + ((size_t)blockIdx.x << 17);
  for (int k = 2; k <= NPAD; k <<= 1) {
    for (int j = k >> 1; j > 0; j >>= 1) {
      for (int i = threadIdx.x; i < NPAD; i += 1024) {
        int ixj = i ^ j;
        if (ixj > i) {
          unsigned long long u = a[i], v = a[ixj];
          bool up = (i & k) == 0;
          if (up ? (u < v) : (u > v)) { a[i] = v; a[ixj] = u; }
        }
      }
      __syncthreads();
    }
  }
}

// ---------------------------------------------------------------------------
// top-6000: decode deltas against on-the-fly anchors (f64 like the reference),
// clip, validity.
// ---------------------------------------------------------------------------
__global__ void build_boxes(const unsigned long long* __restrict__ keys,
                            const float* __restrict__ D, float* __restrict__ B6,
                            float* __restrict__ S6, int* __restrict__ V6)
{
  int gi = blockIdx.x*blockDim.x + threadIdx.x;
  if (gi >= NB*NBOX) return;
  int n = gi / NBOX, i = gi % NBOX;
  unsigned long long key = keys[((size_t)n << 17) + i];
  unsigned int s = (unsigned int)(key >> 32);
  unsigned int bbits = (s >> 31) ? (s ^ 0x80000000u) : ~s;
  float score = __uint_as_float(bbits);
  int b = (int)(~(unsigned int)key);
  int p = b / 3, a = b % 3;
  int y = p / WW, x = p % WW;
  const double wsA[3] = {181.01933598375618, 128.0, 90.50966799187809};
  const double hsA[3] = {90.50966799187809, 128.0, 181.01933598375618};
  double gx = (double)x * 16.0, gy = (double)y * 16.0;
  float ax1 = (float)(gx - wsA[a]*0.5), ay1 = (float)(gy - hsA[a]*0.5);
  float ax2 = (float)(gx + wsA[a]*0.5), ay2 = (float)(gy + hsA[a]*0.5);
  float wa = ax2 - ax1, ha = ay2 - ay1;
  float cxa = ax1 + 0.5f*wa, cya = ay1 + 0.5f*ha;
  const float* d = D + ((size_t)n*NANCH + b)*4;
  float dx = d[0], dy = d[1];
  float dw = fminf(d[2], SCALE_CLAMP), dh = fminf(d[3], SCALE_CLAMP);
  float cx = dx*wa + cxa, cy = dy*ha + cya;
  float pw = expf(dw)*wa, ph = expf(dh)*ha;
  float x1 = fminf(fmaxf(cx - 0.5f*pw, 0.f), (float)(WW*16));
  float y1 = fminf(fmaxf(cy - 0.5f*ph, 0.f), (float)(HH*16));
  float x2 = fminf(fmaxf(cx + 0.5f*pw, 0.f), (float)(WW*16));
  float y2 = fminf(fmaxf(cy + 0.5f*ph, 0.f), (float)(HH*16));
  float* o = B6 + (size_t)gi*4;
  o[0] = x1; o[1] = y1; o[2] = x2; o[3] = y2;
  S6[gi] = score;
  V6[gi] = (x2 - x1 > 0.f) && (y2 - y1 > 0.f);
}

// pairwise IoU suppression bitmask (bits only for j > i)
__global__ __launch_bounds__(64) void nms_mask(const float* __restrict__ B6,
                                               unsigned long long* __restrict__ mask)
{
  __shared__ float cbx[64*4];
  int n = blockIdx.z, rb = blockIdx.y, cblk = blockIdx.x, tt = threadIdx.x;
  int j0 = cblk*64;
  if (j0 + tt < NBOX) {
    const float* bp = B6 + ((size_t)n*NBOX + j0 + tt)*4;
    cbx[tt*4+0] = bp[0]; cbx[tt*4+1] = bp[1]; cbx[tt*4+2] = bp[2]; cbx[tt*4+3] = bp[3];
  }
  __syncthreads();
  int i = rb*64 + tt;
  if (i >= NBOX) return;
  const float* bi = B6 + ((size_t)n*NBOX + i)*4;
  float x1 = bi[0], y1 = bi[1], x2 = bi[2], y2 = bi[3];
  float a1 = (x2 - x1)*(y2 - y1);
  unsigned long long bits = 0ull;
  int jmax = (NBOX - j0 < 64) ? (NBOX - j0) : 64;
  for (int jj = 0; jj < jmax; ++jj) {
    int j = j0 + jj;
    if (j <= i) continue;
    float lx = fmaxf(x1, cbx[jj*4+0]), ly = fmaxf(y1, cbx[jj*4+1]);
    float rx = fminf(x2, cbx[jj*4+2]), ry = fminf(y2, cbx[jj*4+3]);
    float w = fmaxf(rx - lx, 0.f), h = fmaxf(ry - ly, 0.f);
    float inter = w*h;
    float a2 = (cbx[jj*4+2]-cbx[jj*4+0])*(cbx[jj*4+3]-cbx[jj*4+1]);
    float iou = inter > 0.f ? inter / (a1 + a2 - inter) : 0.f;
    if (iou > 0.7f) bits |= (1ull << jj);
  }
  mask[((size_t)n*NBOX + i)*NWRD + cblk] = bits;
}

// sequential greedy scan (invalid boxes pre-suppressed, exact fori_loop semantics)
__global__ __launch_bounds__(128) void nms_scan(const unsigned long long* __restrict__ mask,
                                                const int* __restrict__ V6, int* __restrict__ keep)
{
  __shared__ unsigned long long rem[NWRD];
  __shared__ int flag;
  int n = blockIdx.x, tt = threadIdx.x;
  for (int w = tt; w < NWRD; w += 128) {
    unsigned long long r = 0ull;
    for (int bb = 0; bb < 64; ++bb) {
      int i = w*64 + bb;
      if (i >= NBOX || !V6[n*NBOX + i]) r |= (1ull << bb);
    }
    rem[w] = r;
  }
  __syncthreads();
  for (int i = 0; i < NBOX; ++i) {
    if (tt == 0) flag = (int)((rem[i >> 6] >> (i & 63)) & 1ull);
    __syncthreads();
    if (!flag) {
      const unsigned long long* mr = mask + ((size_t)n*NBOX + i)*NWRD;
      for (int w = tt; w < NWRD; w += 128) rem[w] |= mr[w];
    }
    if (tt == 0) keep[n*NBOX + i] = !flag;
    __syncthreads();
  }
}

__global__ void out_fill(float* __restrict__ out)
{
  int i = blockIdx.x*blockDim.x + threadIdx.x;
  if (i >= NB*POSTK) return;
  float* o = out + (size_t)i*5;
  o[0] = 0.f; o[1] = 0.f; o[2] = 0.f; o[3] = 0.f; o[4] = NEGV;
}

// kept entries are already score-descending -> write in order, cap 1000
__global__ void out_write(const int* __restrict__ keep, const float* __restrict__ B6,
                          const float* __restrict__ S6, float* __restrict__ out)
{
  int n = blockIdx.x;
  if (threadIdx.x != 0) return;
  int rank = 0;
  for (int i = 0; i < NBOX && rank < POSTK; ++i) {
    if (keep[n*NBOX + i]) {
      const float* bp = B6 + ((size_t)n*NBOX + i)*4;
      float* o = out + ((size_t)n*POSTK + rank)*5;
      o[0] = bp[0]; o[1] = bp[1]; o[2] = bp[2]; o[3] = bp[3];
      o[4] = S6[n*NBOX + i];
      ++rank;
    }
  }
}

extern "C" void kernel_launch(void* const* d_in, const int* in_sizes, int n_in,
                              void* d_out, int out_size, void* d_ws, size_t ws_size,
                              hipStream_t stream)
{
  const float* feat  = (const float*)d_in[0];
  const float* wconv = (const float*)d_in[1];
  const float* bconv = (const float*)d_in[2];
  const float* wobj  = (const float*)d_in[3];
  const float* bobj  = (const float*)d_in[4];
  const float* wdel  = (const float*)d_in[5];
  const float* bdel  = (const float*)d_in[6];

  char* ws = (char*)d_ws;
  size_t off = 0;
  auto alloc = [&](size_t bytes) {
    void* p = ws + off;
    off = (off + bytes + 255) & ~(size_t)255;
    return p;
  };
  float* t  = (float*)alloc((size_t)NB*CC*HW*4);            // ~100.7 MB
  float* S  = (float*)alloc((size_t)NB*NANCH*4);            // logits
  float* D  = (float*)alloc((size_t)NB*NANCH*4*4);          // deltas
  unsigned long long* keys = (unsigned long long*)alloc((size_t)NB*NPAD*8);
  float* B6 = (float*)alloc((size_t)NB*NBOX*4*4);
  float* S6 = (float*)alloc((size_t)NB*NBOX*4);
  int*   V6 = (int*)alloc((size_t)NB*NBOX*4);
  unsigned long long* mask = (unsigned long long*)alloc((size_t)NB*NBOX*NWRD*8);
  int* keep = (int*)alloc((size_t)NB*NBOX*4);

  dim3 gc(HW/64, CC/64, NB);
  conv3_wmma<<<gc, 256, 0, stream>>>(feat, wconv, bconv, t);

  dim3 gh(HW/128, NB);
  heads_wmma<<<gh, 256, 0, stream>>>(t, wobj, bobj, wdel, bdel, S, D);

  make_keys<<<(NB*NPAD)/256, 256, 0, stream>>>(S, keys);
  bitonic_sort<<<NB, 1024, 0, stream>>>(keys);
  build_boxes<<<(NB*NBOX + 255)/256, 256, 0, stream>>>(keys, D, B6, S6, V6);

  dim3 gm(NWRD, NWRD, NB);
  nms_mask<<<gm, 64, 0, stream>>>(B6, mask);
  nms_scan<<<NB, 128, 0, stream>>>(mask, V6, keep);

  out_fill<<<(NB*POSTK + 255)/256, 256, 0, stream>>>((float*)d_out);
  out_write<<<NB, 32, 0, stream>>>(keep, B6, S6, (float*)d_out);
}